// my_mse_loss2_68109591379971
// MI455X (gfx1250) — compile-verified
//
#include <hip/hip_runtime.h>
#include <hip/hip_bf16.h>

typedef __attribute__((ext_vector_type(2))) float v2f;
typedef __attribute__((ext_vector_type(8))) float v8f;

#define NBINS     4096      // float bits [30:19]: 8 exp bits + 4 mantissa bits
#define CAP       4096      // candidate capacity (expected boundary ~100 rows)
#define K2_BLOCKS 1024

// ---------------------------------------------------------------------------
// K1: per-row MSE via V_WMMA_F32_16X16X4_F32.
// Each wave handles a 16-row tile. Lane r (r<16) loads elements {0,1} of row r,
// lane r+16 loads elements {2,3}. A-matrix layout (16x4 f32): VGPR0 = K0 (lanes
// 0-15) / K2 (lanes 16-31), VGPR1 = K1 / K3  -> exactly what we loaded.
// B = all 0.25 (layout-invariant), so D[m][n] = row_mse[m] for all n.
// D layout: VGPR j, lanes 0-15 = row_mse[j], lanes 16-31 = row_mse[8+j].
// ---------------------------------------------------------------------------
__global__ void k1_rowmse_wmma(const float* __restrict__ inp,
                               const float* __restrict__ tgt,
                               float* __restrict__ rowmse, long ntiles) {
  const int lane = threadIdx.x & 31;
  const int r    = lane & 15;
  const int h    = lane >> 4;                    // which float2 half of the row
  const long wid    = ((long)blockIdx.x * blockDim.x + threadIdx.x) >> 5;
  const long nwaves = ((long)gridDim.x * blockDim.x) >> 5;
  const float2* in2 = (const float2*)inp;
  const float2* tg2 = (const float2*)tgt;

  for (long tile = wid; tile < ntiles; tile += nwaves) {
    const long row = tile * 16 + r;
    // prefetch a few tiles ahead on this wave's stride
    {
      const long prow = (tile + 4 * nwaves) * 16 + r;
      __builtin_prefetch(in2 + prow * 2 + h, 0, 0);
      __builtin_prefetch(tg2 + prow * 2 + h, 0, 0);
    }
    float2 a = in2[row * 2 + h];
    float2 t = tg2[row * 2 + h];
    float dx = a.x - t.x, dy = a.y - t.y;
    v2f A;  A[0] = dx * dx; A[1] = dy * dy;
    v2f Bm; Bm[0] = 0.25f;  Bm[1] = 0.25f;
    v8f C = {};
    C = __builtin_amdgcn_wmma_f32_16x16x4_f32(
        /*neg_a=*/false, A, /*neg_b=*/false, Bm,
        /*c_mod=*/(short)0, C, /*reuse_a=*/false, /*reuse_b=*/false);
    const long base = tile * 16;
    if (lane == 0) {
      *(float4*)(rowmse + base)     = make_float4(C[0], C[1], C[2], C[3]);
      *(float4*)(rowmse + base + 4) = make_float4(C[4], C[5], C[6], C[7]);
    } else if (lane == 16) {
      *(float4*)(rowmse + base + 8)  = make_float4(C[0], C[1], C[2], C[3]);
      *(float4*)(rowmse + base + 12) = make_float4(C[4], C[5], C[6], C[7]);
    }
  }
}

// Scalar tail for nrows not divisible by 16 (unused for B=4M, kept for safety).
__global__ void k1b_rowmse_tail(const float* __restrict__ inp,
                                const float* __restrict__ tgt,
                                float* __restrict__ rowmse,
                                long start, long nrows) {
  long i = start + blockIdx.x * blockDim.x + threadIdx.x;
  if (i < nrows) {
    float4 a = *(const float4*)(inp + i * 4);
    float4 t = *(const float4*)(tgt + i * 4);
    float d0 = a.x - t.x, d1 = a.y - t.y, d2 = a.z - t.z, d3 = a.w - t.w;
    rowmse[i] = 0.25f * (d0 * d0 + d1 * d1 + d2 * d2 + d3 * d3);
  }
}

// ---------------------------------------------------------------------------
// K2: LDS-privatized histogram of float-bit bins + per-block double partials.
// row_mse >= 0, so the uint32 bit pattern is monotonic in value.
// ---------------------------------------------------------------------------
__global__ void k2_hist_sum(const float* __restrict__ rowmse, long n,
                            int* __restrict__ ghist,
                            double* __restrict__ partials) {
  __shared__ int hist[NBINS];
  __shared__ double red[256];
  for (int b = threadIdx.x; b < NBINS; b += blockDim.x) hist[b] = 0;
  __syncthreads();

  double s = 0.0;
  const long stride = (long)gridDim.x * blockDim.x;
  for (long i = (long)blockIdx.x * blockDim.x + threadIdx.x; i < n; i += stride) {
    float v = rowmse[i];
    unsigned key = __float_as_uint(v);
    atomicAdd(&hist[key >> 19], 1);
    s += (double)v;
  }
  red[threadIdx.x] = s;
  __syncthreads();
  for (int off = blockDim.x >> 1; off > 0; off >>= 1) {
    if ((int)threadIdx.x < off) red[threadIdx.x] += red[threadIdx.x + off];
    __syncthreads();
  }
  if (threadIdx.x == 0) partials[blockIdx.x] = red[0];
  for (int b = threadIdx.x; b < NBINS; b += blockDim.x) {
    int c = hist[b];
    if (c) atomicAdd(&ghist[b], c);
  }
}

// ---------------------------------------------------------------------------
// K3: single thread — deterministic (fixed-order) sum of partials -> mse,
// and histogram scan -> boundary bin T for the n-smallest threshold.
// ---------------------------------------------------------------------------
__global__ void k3_scan(const int* __restrict__ ghist,
                        const double* __restrict__ partials, int nparts,
                        long nrows, const int* __restrict__ topn_p,
                        int* __restrict__ scal, float* __restrict__ out_mse) {
  if (threadIdx.x == 0 && blockIdx.x == 0) {
    double s = 0.0;
    for (int i = 0; i < nparts; ++i) s += partials[i];
    *out_mse = (float)(s / (double)nrows);   // mean of all sq == sum(rowmse)/nrows
    const int topn = *topn_p;
    long cum = 0; int T = NBINS - 1;
    for (int b = 0; b < NBINS; ++b) {
      cum += (long)ghist[b];
      if (cum >= (long)topn) { T = b; break; }
    }
    scal[0] = T;
  }
}

// ---------------------------------------------------------------------------
// K4: collect all rows whose bin <= T as packed (key<<32 | index) candidates.
// ---------------------------------------------------------------------------
__global__ void k4_collect(const float* __restrict__ rowmse, long n,
                           const int* __restrict__ scal,
                           int* __restrict__ counter,
                           unsigned long long* __restrict__ cand) {
  const int T = scal[0];
  const long stride = (long)gridDim.x * blockDim.x;
  for (long i = (long)blockIdx.x * blockDim.x + threadIdx.x; i < n; i += stride) {
    unsigned key = __float_as_uint(rowmse[i]);
    if ((int)(key >> 19) <= T) {
      int pos = atomicAdd(counter, 1);
      if (pos < CAP) cand[pos] = ((unsigned long long)key << 32) | (unsigned)i;
    }
  }
}

// ---------------------------------------------------------------------------
// K5: bitonic sort of CAP candidates in LDS (asc by (value, index) — matches
// top_k tie-breaking), emit first topn indices as float.
// ---------------------------------------------------------------------------
__global__ void k5_sort_emit(const unsigned long long* __restrict__ cand,
                             const int* __restrict__ counter,
                             const int* __restrict__ topn_p,
                             float* __restrict__ out_idx) {
  __shared__ unsigned long long s[CAP];
  int cnt = *counter; if (cnt > CAP) cnt = CAP;
  for (int i = threadIdx.x; i < CAP; i += blockDim.x)
    s[i] = (i < cnt) ? cand[i] : ~0ull;
  __syncthreads();
  for (int k = 2; k <= CAP; k <<= 1) {
    for (int j = k >> 1; j > 0; j >>= 1) {
      for (int t = threadIdx.x; t < CAP; t += blockDim.x) {
        int ixj = t ^ j;
        if (ixj > t) {
          unsigned long long a = s[t], b = s[ixj];
          bool up = ((t & k) == 0);
          if (up ? (a > b) : (a < b)) { s[t] = b; s[ixj] = a; }
        }
      }
      __syncthreads();
    }
  }
  const int topn = *topn_p;
  for (int i = threadIdx.x; i < topn; i += blockDim.x)
    out_idx[i] = (float)(unsigned)(s[i] & 0xffffffffu);
}

// ---------------------------------------------------------------------------
extern "C" void kernel_launch(void* const* d_in, const int* in_sizes, int n_in,
                              void* d_out, int out_size, void* d_ws, size_t ws_size,
                              hipStream_t stream) {
  const float* inp = (const float*)d_in[0];
  const float* tgt = (const float*)d_in[1];
  const int* topn_p = (const int*)d_in[2];       // n=1024 lives on device
  float* out = (float*)d_out;

  const long nrows  = (long)in_sizes[0] / 4;     // D = 4
  const long ntiles = nrows / 16;
  const long tail   = nrows - ntiles * 16;

  // workspace layout
  char* w = (char*)d_ws;
  size_t off = 0;
  float* rowmse = (float*)(w + off);  off += (size_t)nrows * sizeof(float);
  off = (off + 255) & ~(size_t)255;
  int* ghist = (int*)(w + off);       off += NBINS * sizeof(int);
  double* partials = (double*)(w + off); off += K2_BLOCKS * sizeof(double);
  int* scal = (int*)(w + off);        off += 64;         // [0] = threshold bin
  int* counter = (int*)(w + off);     off += 64;
  unsigned long long* cand = (unsigned long long*)(w + off);
  off += (size_t)CAP * sizeof(unsigned long long);
  (void)ws_size; (void)n_in; (void)out_size;

  // per-call zeroing of accumulators (graph-capturable async memsets)
  hipMemsetAsync(ghist, 0, NBINS * sizeof(int), stream);
  hipMemsetAsync(counter, 0, sizeof(int), stream);

  k1_rowmse_wmma<<<1024, 256, 0, stream>>>(inp, tgt, rowmse, ntiles);
  if (tail > 0) {
    k1b_rowmse_tail<<<1, 32, 0, stream>>>(inp, tgt, rowmse, ntiles * 16, nrows);
  }
  k2_hist_sum<<<K2_BLOCKS, 256, 0, stream>>>(rowmse, nrows, ghist, partials);
  k3_scan<<<1, 32, 0, stream>>>(ghist, partials, K2_BLOCKS, nrows, topn_p, scal, out);
  k4_collect<<<1024, 256, 0, stream>>>(rowmse, nrows, scal, counter, cand);
  k5_sort_emit<<<1, 1024, 0, stream>>>(cand, counter, topn_p, out + 1);
}